// GraphSAGE_20057497272825
// MI455X (gfx1250) — compile-verified
//
#include <hip/hip_runtime.h>
#include <hip/hip_bf16.h>

#define N_NODES 10000
#define N_EDGES 640000
#define IN_DIM  128
#define HID_DIM 256
#define OUT_DIM 128

typedef __attribute__((ext_vector_type(2))) float v2f;
typedef __attribute__((ext_vector_type(8))) float v8f;

// ---------------- utility kernels ----------------

__global__ void zero_f32(float* __restrict__ p, int n) {
    int i = blockIdx.x * blockDim.x + threadIdx.x;
    if (i < n) p[i] = 0.0f;
}

__global__ void degree_kernel(const int* __restrict__ dst, float* __restrict__ deg, int nE) {
    int e = blockIdx.x * blockDim.x + threadIdx.x;
    if (e < nE) atomicAdd(&deg[dst[e]], 1.0f);
}

__global__ void invdeg_kernel(const float* __restrict__ deg, float* __restrict__ invd, int n) {
    int i = blockIdx.x * blockDim.x + threadIdx.x;
    if (i < n) invd[i] = 1.0f / fmaxf(deg[i], 1.0f);
}

// Scatter-sum of source-node features into per-destination accumulators.
// One thread per (edge, 4-float chunk). Consecutive threads cover consecutive
// chunks of the same edge -> coalesced gather from feat[src], coalesced-ish
// atomic adds into msg[dst]. Atomics resolve in L2 (f32 add units).
template<int D4>  // D/4
__global__ void scatter_kernel(const float* __restrict__ feat,
                               const int* __restrict__ src,
                               const int* __restrict__ dst,
                               float* __restrict__ msg, int nE) {
    int i = blockIdx.x * blockDim.x + threadIdx.x;
    int e = i / D4;
    int g = i % D4;
    if (e >= nE) return;
    int s = src[e];
    int d = dst[e];
    const float4 v = ((const float4*)(feat + (long long)s * (D4 * 4)))[g];
    float* o = msg + (long long)d * (D4 * 4) + g * 4;
    atomicAdd(o + 0, v.x);
    atomicAdd(o + 1, v.y);
    atomicAdd(o + 2, v.z);
    atomicAdd(o + 3, v.w);
}

// ---------------- fused SAGE GEMM ----------------
// out = act( (msg * invdeg_row) @ Wl  +  feat @ Wr  + bias )
// M is a multiple of 16 (10000 = 625*16); N, K multiples of 16/4.
// One wave32 computes a 16x16 f32 tile via V_WMMA_F32_16X16X4_F32 (full fp32
// precision; GEMM is compute-trivial vs the scatter phase, so no reason to
// drop to bf16). 8 waves per 256-thread block, no divergence -> EXEC all-1s
// as WMMA requires.
//
// ISA §7.12.2 f32 layouts (wave32):
//   A 16x4 : lane m = lane&15, a[j] = A[m][2*(lane>>4)+j]   (contiguous pair)
//   B 4x16 : lane n = lane&15, b[j] = B[2*(lane>>4)+j][n]
//   C/D    : lane n = lane&15, c[r] = C[8*(lane>>4)+r][n]
template<int K, int N, bool RELU>
__global__ __launch_bounds__(256) void sage_gemm(
    const float* __restrict__ msg, const float* __restrict__ invdeg,
    const float* __restrict__ feat,
    const float* __restrict__ Wl, const float* __restrict__ Wr,
    const float* __restrict__ bias, float* __restrict__ out) {

    const int lane = threadIdx.x & 31;
    const int wave = threadIdx.x >> 5;
    const int t    = blockIdx.x * 8 + wave;     // linear tile id
    const int NT   = N / 16;
    const int mt   = t / NT;
    const int nt   = t % NT;

    const int mrow  = mt * 16 + (lane & 15);    // A row for this lane
    const int ncol  = nt * 16 + (lane & 15);    // B/C/D column for this lane
    const int khalf = (lane >> 4) * 2;          // 0 or 2

    v8f acc;
    const float bv = bias[ncol];
    #pragma unroll
    for (int r = 0; r < 8; ++r) acc[r] = bv;

    const float  invd = invdeg[mrow];
    const float* arow = msg  + (long long)mrow * K;
    const float* frow = feat + (long long)mrow * K;

    // Pass 1: mean-aggregated neighbors (normalization fused into A load)
    #pragma unroll 4
    for (int k0 = 0; k0 < K; k0 += 4) {
        const float2 av = *(const float2*)(arow + k0 + khalf);
        v2f a, b;
        a[0] = av.x * invd;
        a[1] = av.y * invd;
        b[0] = Wl[(long long)(k0 + khalf    ) * N + ncol];
        b[1] = Wl[(long long)(k0 + khalf + 1) * N + ncol];
        acc = __builtin_amdgcn_wmma_f32_16x16x4_f32(false, a, false, b,
                                                    (short)0, acc, false, false);
    }
    // Pass 2: self features
    #pragma unroll 4
    for (int k0 = 0; k0 < K; k0 += 4) {
        const float2 fv = *(const float2*)(frow + k0 + khalf);
        v2f a, b;
        a[0] = fv.x;
        a[1] = fv.y;
        b[0] = Wr[(long long)(k0 + khalf    ) * N + ncol];
        b[1] = Wr[(long long)(k0 + khalf + 1) * N + ncol];
        acc = __builtin_amdgcn_wmma_f32_16x16x4_f32(false, a, false, b,
                                                    (short)0, acc, false, false);
    }

    const int mbase = mt * 16 + (lane >> 4) * 8;
    #pragma unroll
    for (int r = 0; r < 8; ++r) {
        float v = acc[r];
        if (RELU) v = fmaxf(v, 0.0f);
        out[(long long)(mbase + r) * N + ncol] = v;
    }
}

// ---------------- launch ----------------

extern "C" void kernel_launch(void* const* d_in, const int* in_sizes, int n_in,
                              void* d_out, int out_size, void* d_ws, size_t ws_size,
                              hipStream_t stream) {
    const float* x   = (const float*)d_in[0];
    const int*   ei  = (const int*)d_in[1];    // [2, E]
    const float* W1l = (const float*)d_in[2];
    const float* b1  = (const float*)d_in[3];
    const float* W1r = (const float*)d_in[4];
    const float* W2l = (const float*)d_in[5];
    const float* b2  = (const float*)d_in[6];
    const float* W2r = (const float*)d_in[7];
    float* out = (float*)d_out;

    const int* src = ei;
    const int* dst = ei + N_EDGES;

    // Workspace layout (floats). Buffers needing per-call zeroing are placed
    // first and contiguous so a single zero kernel clears them.
    const size_t PAD_N = 10240;  // node-count padded to 256-multiple
    float* deg  = (float*)d_ws;                                  // PAD_N
    float* msg1 = deg  + PAD_N;                                  // N*IN_DIM
    float* msg2 = msg1 + (size_t)N_NODES * IN_DIM;               // N*HID_DIM
    float* invd = msg2 + (size_t)N_NODES * HID_DIM;              // PAD_N
    float* h1   = invd + PAD_N;                                  // N*HID_DIM

    const int nzero = (int)(PAD_N + (size_t)N_NODES * IN_DIM
                                  + (size_t)N_NODES * HID_DIM);  // 3,850,240

    // 1) clear accumulators (harness does not re-poison between replays)
    zero_f32<<<(nzero + 255) / 256, 256, 0, stream>>>(deg, nzero);

    // 2) degree counts + reciprocal
    degree_kernel<<<(N_EDGES + 255) / 256, 256, 0, stream>>>(dst, deg, N_EDGES);
    invdeg_kernel<<<(N_NODES + 255) / 256, 256, 0, stream>>>(deg, invd, N_NODES);

    // 3) layer-1 scatter: msg1[dst] += x[src]   (D = 128)
    {
        const int total = N_EDGES * (IN_DIM / 4);
        scatter_kernel<IN_DIM / 4><<<(total + 255) / 256, 256, 0, stream>>>(
            x, src, dst, msg1, N_EDGES);
    }

    // 4) h1 = relu( (msg1/deg) @ W1_l + x @ W1_r + b1 )   M=10000 K=128 N=256
    {
        const int tiles = (N_NODES / 16) * (HID_DIM / 16);   // 625*16 = 10000
        sage_gemm<IN_DIM, HID_DIM, true><<<tiles / 8, 256, 0, stream>>>(
            msg1, invd, x, W1l, W1r, b1, h1);
    }

    // 5) layer-2 scatter: msg2[dst] += h1[src]  (D = 256)
    {
        const int total = N_EDGES * (HID_DIM / 4);
        scatter_kernel<HID_DIM / 4><<<(total + 255) / 256, 256, 0, stream>>>(
            h1, src, dst, msg2, N_EDGES);
    }

    // 6) out = (msg2/deg) @ W2_l + h1 @ W2_r + b2   M=10000 K=256 N=128
    {
        const int tiles = (N_NODES / 16) * (OUT_DIM / 16);   // 625*8 = 5000
        sage_gemm<HID_DIM, OUT_DIM, false><<<tiles / 8, 256, 0, stream>>>(
            msg2, invd, h1, W2l, W2r, b2, out);
    }
}